// HadamardRotation_14190571946379
// MI455X (gfx1250) — compile-verified
//
#include <hip/hip_runtime.h>

// CDNA5 / gfx1250 FWHT-4096 via WMMA: H4096 = H16 (x) H16 (x) H16.
// One wave32 per row; 3 passes of 16x16 GEMM tiles using V_WMMA_F32_16X16X4_F32.

typedef __attribute__((ext_vector_type(2))) float v2f;
typedef __attribute__((ext_vector_type(4))) float v4f;
typedef __attribute__((ext_vector_type(8))) float v8f;

#define DIM   4096
#define SJ    19            // LDS DWORD stride between consecutive j (mid-axis) entries
#define SI    308           // LDS DWORD stride between consecutive i entries (16*19 + 4)
#define LDSDW (16 * SI)     // 4928 DWORDs = 19712 bytes per row buffer

__global__ __launch_bounds__(32)
void fwht4096_wmma(const float* __restrict__ in, float* __restrict__ out, int nrows)
{
    __shared__ float buf[LDSDW];

    const int lane = threadIdx.x & 31;
    const int n15  = lane & 15;   // column index within a 16-wide tile
    const int hi   = lane >> 4;   // upper half-wave flag (K/M offset per ISA layouts)

    const int row = blockIdx.x;
    if (row >= nrows) return;     // uniform branch: EXEC stays all-ones for WMMA

    // A operands for the four K=4 chunks of H16.
    // f32 16x4 A layout: lanes 0-15 hold K={0,1}, lanes 16-31 hold K={2,3} (M = lane&15).
    // Chunk q covers logical K = 4q..4q+3, so this lane supplies K = 4q+2*hi, 4q+1+2*hi.
    // H16[m][k] = (-1)^popcount(m & k).  Pass-3 A is pre-scaled by 1/64 (exact).
    v2f A1[4], A3[4];
#pragma unroll
    for (int q = 0; q < 4; ++q) {
        const int k0 = 4 * q + 2 * hi;
        const float s0 = (__popc(n15 & k0)       & 1) ? -1.0f : 1.0f;
        const float s1 = (__popc(n15 & (k0 + 1)) & 1) ? -1.0f : 1.0f;
        A1[q] = (v2f){s0, s1};
        A3[q] = (v2f){s0 * 0.015625f, s1 * 0.015625f};
    }

    const float* __restrict__ src = in  + (size_t)row * DIM;
    float*       __restrict__ dst = out + (size_t)row * DIM;

    // ---------------- Pass 1: transform along k (stride-1 axis) ----------------
    // Tile t = i.  Column n = j = lane&15.  Fiber = 16 contiguous floats in global.
    // B 4x16 layout: lanes 0-15 hold rows K={0,1}, lanes 16-31 rows K={2,3} -> b64 loads,
    // the two half-waves together cover each 64B chunk => fully coalesced.
#pragma unroll 1
    for (int t = 0; t < 16; ++t) {
        const v2f* fb = (const v2f*)(src + t * 256 + n15 * 16);
        v2f B[4];
#pragma unroll
        for (int q = 0; q < 4; ++q)
            B[q] = __builtin_nontemporal_load(&fb[2 * q + hi]);   // rows 4q+2hi, 4q+1+2hi

        v8f acc = {0.f,0.f,0.f,0.f,0.f,0.f,0.f,0.f};
#pragma unroll
        for (int q = 0; q < 4; ++q)
            acc = __builtin_amdgcn_wmma_f32_16x16x4_f32(
                      false, A1[q], false, B[q], (short)0, acc, false, false);

        // D[m][n]: m = a = r + 8*hi, n = j.  Store Y1[i=t][j][a].
        const int base = t * SI + n15 * SJ + 8 * hi;
#pragma unroll
        for (int r = 0; r < 8; ++r)
            buf[base + r] = acc[r];
    }

    // ---------------- Pass 2: transform along j ----------------
    // Tile u = i.  Column n = a = lane&15.  Fiber rows = j (LDS stride SJ).
#pragma unroll 1
    for (int u = 0; u < 16; ++u) {
        const int tb = u * SI + n15;
        v2f B[4];
#pragma unroll
        for (int q = 0; q < 4; ++q) {
            const int j0 = 4 * q + 2 * hi;
            B[q][0] = buf[tb + j0 * SJ];
            B[q][1] = buf[tb + (j0 + 1) * SJ];
        }
        v8f acc = {0.f,0.f,0.f,0.f,0.f,0.f,0.f,0.f};
#pragma unroll
        for (int q = 0; q < 4; ++q)
            acc = __builtin_amdgcn_wmma_f32_16x16x4_f32(
                      false, A1[q], false, B[q], (short)0, acc, false, false);

        // m = b: write Y2[i=u][b][a] back in place (reads of this tile already done).
#pragma unroll
        for (int r = 0; r < 8; ++r)
            buf[tb + (r + 8 * hi) * SJ] = acc[r];
    }

    // ---------------- Pass 3: transform along i, scaled by 1/64 ----------------
    // Tile w = b.  Column n = a = lane&15.  Fiber rows = i (LDS stride SI).
#pragma unroll 1
    for (int w = 0; w < 16; ++w) {
        const int tb = w * SJ + n15;
        v2f B[4];
#pragma unroll
        for (int q = 0; q < 4; ++q) {
            const int i0 = 4 * q + 2 * hi;
            B[q][0] = buf[tb + i0 * SI];
            B[q][1] = buf[tb + (i0 + 1) * SI];
        }
        v8f acc = {0.f,0.f,0.f,0.f,0.f,0.f,0.f,0.f};
#pragma unroll
        for (int q = 0; q < 4; ++q)
            acc = __builtin_amdgcn_wmma_f32_16x16x4_f32(
                      false, A3[q], false, B[q], (short)0, acc, false, false);

#pragma unroll
        for (int r = 0; r < 8; ++r)
            buf[tb + (r + 8 * hi) * SI] = acc[r];
    }

    // ---------------- Coalesced, nontemporal write-back ----------------
    // y[c*256 + b*16 + a] lives at LDS DWORD c*SI + b*SJ + a.
#pragma unroll 1
    for (int it = 0; it < 32; ++it) {
        const int idx = (it * 32 + lane) * 4;                 // 16B-aligned, a&3 == 0
        const int o = (idx >> 8) * SI + ((idx >> 4) & 15) * SJ + (idx & 15);
        v4f v = { buf[o], buf[o + 1], buf[o + 2], buf[o + 3] };
        __builtin_nontemporal_store(v, (v4f*)(dst + idx));
    }
}

extern "C" void kernel_launch(void* const* d_in, const int* in_sizes, int n_in,
                              void* d_out, int out_size, void* d_ws, size_t ws_size,
                              hipStream_t stream)
{
    const float* x = (const float*)d_in[0];
    float*       y = (float*)d_out;
    const int nrows = in_sizes[0] / DIM;   // 4 * 4096 = 16384 rows
    fwht4096_wmma<<<nrows, 32, 0, stream>>>(x, y, nrows);
}